// GNNLayer_11476152614989
// MI455X (gfx1250) — compile-verified
//
#include <hip/hip_runtime.h>
#include <math.h>

// ---------------------------------------------------------------------------
// GNN layer (temporal KG attention + segment-sum + output projection), fp32.
// - All per-edge GEMVs factored into precomputed tables (node/rel/query/time).
// - Edge kernel: wave32 per edge, gather + shfl-reduce + atomic scatter.
//   Atomics halved: scatter alpha*msg into agg and msg into agg_tot;
//   agg_s = agg_tot - agg is formed inside the second WMMA GEMM's A path.
// - Big GEMMs use V_WMMA_F32_16X16X4_F32 with B staged in LDS, 4 row strips
//   per block to amortize the 64KB LDS stage.
// ---------------------------------------------------------------------------

typedef __attribute__((ext_vector_type(2))) float v2f;
typedef __attribute__((ext_vector_type(8))) float v8f;

#define IN_DIM   128
#define OUT_DIM  128
#define ATTN_DIM 64
#define T_TAB    366   // tau in [-1, 365]; tau<0 mapped to q_tau

// ---------------- WMMA fp32 GEMM: C[M x N] = (A0[-A1])[M x 128] @ B[128 x N]
// blockDim = (N/16)*32; wave w computes columns [16w,16w+16).
// Each block processes STRIPS row strips of 16 to amortize the B LDS stage.
template <int N, bool DIFF>
__global__ void wmma_gemm_k128(const float* __restrict__ A0,
                               const float* __restrict__ A1,
                               const float* __restrict__ B,
                               float* __restrict__ C, int M) {
    constexpr int K = 128;
    constexpr int STRIPS = 4;
    __shared__ float Blds[K * N];

    // Cooperative load of B (128 x N) into LDS, float4-vectorized.
    const float4* B4 = reinterpret_cast<const float4*>(B);
    float4* L4 = reinterpret_cast<float4*>(Blds);
    for (int i = threadIdx.x; i < K * N / 4; i += blockDim.x) L4[i] = B4[i];
    __syncthreads();

    const int wave    = threadIdx.x >> 5;
    const int lane    = threadIdx.x & 31;
    const int colBase = wave * 16;
    const int mn      = lane & 15;          // A: M index ; B: N index
    const int khalf   = (lane >> 4) * 2;    // lanes 16-31 hold K+2,K+3
    const int rbase   = (lane >> 4) * 8;    // C rows: lanes 16-31 -> M=8..15
    const int col     = colBase + mn;

    for (int s = 0; s < STRIPS; ++s) {
        const int row0 = (blockIdx.x * STRIPS + s) * 16;
        if (row0 >= M) break;

        int arow = row0 + mn;
        if (arow >= M) arow = M - 1;        // clamp loads (stores guarded)
        const float* __restrict__ Ap0 = A0 + (size_t)arow * K;
        const float* __restrict__ Ap1 = A1 + (size_t)arow * K;

        v8f c = {0.f, 0.f, 0.f, 0.f, 0.f, 0.f, 0.f, 0.f};
#pragma unroll
        for (int k = 0; k < K; k += 4) {
            const int ko = k + khalf;
            v2f a;
            if (DIFF) {
                a[0] = Ap0[ko]     - Ap1[ko];
                a[1] = Ap0[ko + 1] - Ap1[ko + 1];
            } else {
                a[0] = Ap0[ko];
                a[1] = Ap0[ko + 1];
            }
            v2f b;
            b[0] = Blds[ ko      * N + colBase + mn];
            b[1] = Blds[(ko + 1) * N + colBase + mn];
            // (neg_a, A, neg_b, B, c_mod, C, reuse_a, reuse_b)
            c = __builtin_amdgcn_wmma_f32_16x16x4_f32(false, a, false, b,
                                                      (short)0, c, false, false);
        }

#pragma unroll
        for (int r = 0; r < 8; ++r) {
            const int row = row0 + rbase + r;
            if (row < M) C[(size_t)row * N + col] = c[r];
        }
    }
}

// ---------------- time-feature table: htab[t][d], t in [0,366) -------------
__global__ void htau_kernel(const float* __restrict__ w1, const float* __restrict__ b1,
                            const float* __restrict__ w2, const float* __restrict__ b2,
                            const int* __restrict__ qtau, float* __restrict__ htab) {
    const int i = blockIdx.x * blockDim.x + threadIdx.x;
    if (i >= T_TAB * IN_DIM) return;
    const int t = i / IN_DIM, d = i % IN_DIM;
    const float delta = (float)(t - qtau[0]);
    htab[i] = w1[d] * delta + b1[d] + sinf(w2[d] * delta + b2[d]);
}

// ---------------- tiny projections: Y[rows x 64] = X[rows x 128] @ W -------
__global__ void small_proj_kernel(const float* __restrict__ X, const float* __restrict__ W,
                                  float* __restrict__ Y, int rows) {
    const int i = blockIdx.x * blockDim.x + threadIdx.x;
    if (i >= rows * ATTN_DIM) return;
    const int r = i / ATTN_DIM, j = i % ATTN_DIM;
    const float* __restrict__ x = X + (size_t)r * IN_DIM;
    float acc = 0.f;
#pragma unroll 4
    for (int d = 0; d < IN_DIM; ++d) acc += x[d] * W[d * ATTN_DIM + j];
    Y[i] = acc;
}

__global__ void wqrq_kernel(const int* __restrict__ q_rel, const float* __restrict__ rela,
                            const float* __restrict__ W, const float* __restrict__ bvec,
                            float* __restrict__ Y, int nq) {
    const int i = blockIdx.x * blockDim.x + threadIdx.x;
    if (i >= nq * ATTN_DIM) return;
    const int q = i / ATTN_DIM, j = i % ATTN_DIM;
    const float* __restrict__ x = rela + (size_t)q_rel[q] * IN_DIM;
    float acc = bvec[j];
    for (int d = 0; d < IN_DIM; ++d) acc += x[d] * W[d * ATTN_DIM + j];
    Y[i] = acc;
}

// ---------------- edge kernel: one wave32 per edge -------------------------
// attn(64) = relu(WsH[sub] + WrR[rel] + WqrQ[r_idx] + WtauT[t])  (2 dims/lane)
// alpha    = sigmoid(attn . w_alpha + b)                         (shfl_xor)
// message(128) = hidden[sub] + rela[rel] + htau[t]               (4 dims/lane)
// scatter: agg[obj]     += alpha * message
//          agg_tot[obj] += message          (agg_s formed later as tot - agg)
__global__ void edge_kernel(const int* __restrict__ edges,
                            const float* __restrict__ hidden,
                            const float* __restrict__ rela,
                            const float* __restrict__ WsH,
                            const float* __restrict__ WrR,
                            const float* __restrict__ WqrQ,
                            const float* __restrict__ WtauT,
                            const float* __restrict__ htab,
                            const float* __restrict__ walpha,
                            const float* __restrict__ walpha_b,
                            const int* __restrict__ qtau,
                            float* __restrict__ agg,
                            float* __restrict__ agg_tot,
                            int n_edge) {
    const int e = blockIdx.x * (blockDim.x >> 5) + (threadIdx.x >> 5);
    const int lane = threadIdx.x & 31;
    if (e >= n_edge) return;

    const int* __restrict__ E = edges + (size_t)e * 7;
    const int r_idx = E[0];
    const int rel   = E[2];
    const int tauv  = E[4];
    const int sub   = E[5];
    const int obj   = E[6];
    const int t     = (tauv >= 0) ? tauv : qtau[0];

    // ---- attention score (64 dims, 2 per lane) ----
    const int j = lane * 2;
    float a0 = WsH[(size_t)sub * ATTN_DIM + j]     + WrR[(size_t)rel * ATTN_DIM + j]
             + WqrQ[(size_t)r_idx * ATTN_DIM + j]  + WtauT[(size_t)t * ATTN_DIM + j];
    float a1 = WsH[(size_t)sub * ATTN_DIM + j + 1] + WrR[(size_t)rel * ATTN_DIM + j + 1]
             + WqrQ[(size_t)r_idx * ATTN_DIM + j + 1] + WtauT[(size_t)t * ATTN_DIM + j + 1];
    a0 = fmaxf(a0, 0.f);
    a1 = fmaxf(a1, 0.f);
    float s = a0 * walpha[j] + a1 * walpha[j + 1];
#pragma unroll
    for (int off = 16; off > 0; off >>= 1) s += __shfl_xor(s, off, 32);
    const float alpha = 1.f / (1.f + expf(-(s + walpha_b[0])));

    // ---- message (128 dims, 4 per lane) + scatter-add ----
    const int d = lane * 4;
    const float4 hsv = *reinterpret_cast<const float4*>(hidden + (size_t)sub * IN_DIM + d);
    const float4 hrv = *reinterpret_cast<const float4*>(rela   + (size_t)rel * IN_DIM + d);
    const float4 htv = *reinterpret_cast<const float4*>(htab   + (size_t)t   * IN_DIM + d);
    const float m0 = hsv.x + hrv.x + htv.x;
    const float m1 = hsv.y + hrv.y + htv.y;
    const float m2 = hsv.z + hrv.z + htv.z;
    const float m3 = hsv.w + hrv.w + htv.w;

    float* __restrict__ pa = agg     + (size_t)obj * IN_DIM + d;
    float* __restrict__ pt = agg_tot + (size_t)obj * IN_DIM + d;
    atomicAdd(pa + 0, alpha * m0);
    atomicAdd(pa + 1, alpha * m1);
    atomicAdd(pa + 2, alpha * m2);
    atomicAdd(pa + 3, alpha * m3);
    atomicAdd(pt + 0, m0);
    atomicAdd(pt + 1, m1);
    atomicAdd(pt + 2, m2);
    atomicAdd(pt + 3, m3);
}

// ---------------------------------------------------------------------------
extern "C" void kernel_launch(void* const* d_in, const int* in_sizes, int n_in,
                              void* d_out, int out_size, void* d_ws, size_t ws_size,
                              hipStream_t stream) {
    // setup_inputs() order:
    //  0 q_sub(64 i32)  1 q_rel(64 i32)  2 q_tau(1 i32)  3 hidden(N*128 f32)
    //  4 edges(E*7 i32) 5 n_node(1 i32)  6 old_nodes_new_idx(1000 i32)
    //  7 rela_embed(401*128 f32) 8 Ws_w 9 Wr_w 10 Wqr_w 11 Wqr_b 12 Wtau_w
    // 13 w_alpha_w(64) 14 w_alpha_b(1) 15 W_h_w(128*128) 16 W_h_s_w(128*128)
    // 17 weight_t1 18 bias_t1 19 weight_t2 20 bias_t2
    const int*   q_rel    = (const int*)  d_in[1];
    const int*   q_tau    = (const int*)  d_in[2];
    const float* hidden   = (const float*)d_in[3];
    const int*   edges    = (const int*)  d_in[4];
    const float* rela     = (const float*)d_in[7];
    const float* Ws_w     = (const float*)d_in[8];
    const float* Wr_w     = (const float*)d_in[9];
    const float* Wqr_w    = (const float*)d_in[10];
    const float* Wqr_b    = (const float*)d_in[11];
    const float* Wtau_w   = (const float*)d_in[12];
    const float* walpha   = (const float*)d_in[13];
    const float* walpha_b = (const float*)d_in[14];
    const float* W_h_w    = (const float*)d_in[15];
    const float* W_h_s_w  = (const float*)d_in[16];
    const float* wt1      = (const float*)d_in[17];
    const float* bt1      = (const float*)d_in[18];
    const float* wt2      = (const float*)d_in[19];
    const float* bt2      = (const float*)d_in[20];

    const int n_node = in_sizes[3] / IN_DIM;   // 100000
    const int n_edge = in_sizes[4] / 7;        // 500000
    const int n_relr = in_sizes[7] / IN_DIM;   // 401
    const int n_q    = in_sizes[1];            // 64

    // Workspace layout (floats); all sub-arrays stay 16B-aligned.
    float* ws      = (float*)d_ws;
    float* agg     = ws;                                    // n_node*128
    float* agg_tot = agg     + (size_t)n_node * IN_DIM;     // n_node*128
    float* WsH     = agg_tot + (size_t)n_node * IN_DIM;     // n_node*64
    float* WrR     = WsH     + (size_t)n_node * ATTN_DIM;   // n_relr*64
    float* WqrQ    = WrR     + (size_t)n_relr * ATTN_DIM;   // n_q*64
    float* htab    = WqrQ    + (size_t)n_q * ATTN_DIM;      // 366*128
    float* WtauT   = htab    + (size_t)T_TAB * IN_DIM;      // 366*64

    float* out0 = (float*)d_out;                            // hidden_new
    float* out1 = out0 + (size_t)n_node * OUT_DIM;          // hidden_new_s

    // 1) zero the segment-sum accumulators (workspace is poisoned by harness)
    hipMemsetAsync(agg, 0, (size_t)2 * n_node * IN_DIM * sizeof(float), stream);

    // 2) small precomputed tables
    {
        int n = T_TAB * IN_DIM;
        htau_kernel<<<(n + 255) / 256, 256, 0, stream>>>(wt1, bt1, wt2, bt2, q_tau, htab);
    }
    {
        int n = T_TAB * ATTN_DIM;
        small_proj_kernel<<<(n + 255) / 256, 256, 0, stream>>>(htab, Wtau_w, WtauT, T_TAB);
    }
    {
        int n = n_relr * ATTN_DIM;
        small_proj_kernel<<<(n + 255) / 256, 256, 0, stream>>>(rela, Wr_w, WrR, n_relr);
    }
    {
        int n = n_q * ATTN_DIM;
        wqrq_kernel<<<(n + 255) / 256, 256, 0, stream>>>(q_rel, rela, Wqr_w, Wqr_b, WqrQ, n_q);
    }

    // 3) node projection via WMMA: WsH = hidden @ Ws_w  (100k x 128 @ 128 x 64)
    wmma_gemm_k128<ATTN_DIM, false>
        <<<(n_node + 63) / 64, (ATTN_DIM / 16) * 32, 0, stream>>>(
            hidden, hidden, Ws_w, WsH, n_node);

    // 4) per-edge attention + message + atomic segment-sum (wave32 per edge)
    edge_kernel<<<(n_edge + 7) / 8, 256, 0, stream>>>(
        edges, hidden, rela, WsH, WrR, WqrQ, WtauT, htab,
        walpha, walpha_b, q_tau, agg, agg_tot, n_edge);

    // 5) output projections via WMMA (100k x 128 @ 128 x 128, twice)
    //    out0 = agg @ W_h_w ; out1 = (agg_tot - agg) @ W_h_s_w
    wmma_gemm_k128<OUT_DIM, false>
        <<<(n_node + 63) / 64, (OUT_DIM / 16) * 32, 0, stream>>>(
            agg, agg, W_h_w, out0, n_node);
    wmma_gemm_k128<OUT_DIM, true>
        <<<(n_node + 63) / 64, (OUT_DIM / 16) * 32, 0, stream>>>(
            agg_tot, agg, W_h_s_w, out1, n_node);
}